// TSGM_44744969289764
// MI455X (gfx1250) — compile-verified
//
#include <hip/hip_runtime.h>
#include <math.h>

// CDNA5 / gfx1250 fused TSGM forward.
// One wave32 per 8x8 attention window; v_wmma_f32_16x16x4_f32 for P @ V.

typedef float v2f __attribute__((ext_vector_type(2)));
typedef float v8f __attribute__((ext_vector_type(8)));

#define BATCH 16
#define CIN   3
#define HIN   512
#define WIN_  512
#define EC    12      // C*4
#define WAVES_PER_BLOCK 8
#define TOTAL_WINDOWS (16 * 32 * 32)   // B * (256/8) * (256/8) = 16384
#define PI_F 3.14159265358979323846f

__global__ __launch_bounds__(256) void tsgm_fused_kernel(
    const float* __restrict__ x1, const float* __restrict__ x2,
    const float* __restrict__ t1, const float* __restrict__ t2,
    const float* __restrict__ Wq, const float* __restrict__ bq,
    const float* __restrict__ Wk, const float* __restrict__ bk,
    const float* __restrict__ Wv, const float* __restrict__ bv,
    const float* __restrict__ Wp, const float* __restrict__ bp,
    const float* __restrict__ gamma,
    const float* __restrict__ Wt, const float* __restrict__ bt,
    float* __restrict__ out)
{
    __shared__ float sWv[EC * EC];
    __shared__ float sWq[EC], sWk[EC], sbv[EC], sWp_[EC], sbp[EC];
    __shared__ float sFeat[2][EC];
    __shared__ float sScal[3];                         // bq, bk, gamma
    __shared__ float sQ[WAVES_PER_BLOCK][64];
    __shared__ float sK[WAVES_PER_BLOCK][64];
    __shared__ float sS[WAVES_PER_BLOCK][16];
    __shared__ float sV[WAVES_PER_BLOCK][64 * 16];     // V, cols 12..15 zero-padded

    const int tid  = threadIdx.x;
    const int lane = tid & 31;
    const int wave = tid >> 5;

    // ---------------- stage small parameters / time embeddings ----------------
    for (int i = tid; i < EC * EC; i += 256) sWv[i] = Wv[i];
    if (tid < EC) {
        sWq[tid] = Wq[tid]; sWk[tid] = Wk[tid]; sbv[tid] = bv[tid];
        sWp_[tid] = Wp[tid]; sbp[tid] = bp[tid];
    }
    if (tid < 2 * EC) {
        const int which = tid / EC;
        const int c = tid - which * EC;
        const float t = which ? t2[0] : t1[0];
        float acc = bt[c];
        #pragma unroll
        for (int j = 0; j < 8; ++j) {
            const float ang = t * (float)(j + 1) * PI_F;
            acc += sinf(ang) * Wt[c * 16 + j] + cosf(ang) * Wt[c * 16 + 8 + j];
        }
        sFeat[which][c] = acc;
    }
    if (tid == 0) { sScal[0] = bq[0]; sScal[1] = bk[0]; sScal[2] = gamma[0]; }
    __syncthreads();

    // ---------------- window identification ----------------
    const int wid = blockIdx.x * WAVES_PER_BLOCK + wave;   // 0..16383
    const int b   = wid >> 10;
    const int rem = wid & 1023;
    const int wh  = rem >> 5;
    const int ww  = rem & 31;
    const int baseRow = wh * 16;
    const int baseCol = ww * 16;

    const float* x1b = x1 + (size_t)b * (CIN * HIN * WIN_);
    const float* x2b = x2 + (size_t)b * (CIN * HIN * WIN_);

    // ---------------- per-token gather + q/k/v (2 tokens per lane) ----------------
    float kown0 = 0.f, kown1 = 0.f;
    #pragma unroll
    for (int tt = 0; tt < 2; ++tt) {
        const int n  = lane + tt * 32;
        const int ty = n >> 3, tx = n & 7;
        const int r0 = 4 * (ty >> 1) + (ty & 1);
        const int s0 = 4 * (tx >> 1) + (tx & 1);
        float f1v[EC], f2v[EC];
        #pragma unroll
        for (int c4 = 0; c4 < EC; ++c4) {
            const int c  = c4 >> 2, kk = c4 & 3;
            const int rr = baseRow + r0 + 2 * (kk >> 1);
            const int cc = baseCol + s0 + ((kk & 1) << 1);
            const size_t idx = ((size_t)c * HIN + rr) * WIN_ + cc;
            f1v[c4] = x1b[idx] + sFeat[0][c4];
            f2v[c4] = x2b[idx] + sFeat[1][c4];
        }
        float qv = sScal[0], kv = sScal[1];
        #pragma unroll
        for (int c4 = 0; c4 < EC; ++c4) {
            qv = fmaf(sWq[c4], f1v[c4], qv);
            kv = fmaf(sWk[c4], f2v[c4], kv);
        }
        sQ[wave][n] = qv;
        sK[wave][n] = kv;
        if (tt == 0) kown0 = kv; else kown1 = kv;
        #pragma unroll
        for (int co = 0; co < EC; ++co) {
            float acc = sbv[co];
            #pragma unroll
            for (int ci = 0; ci < EC; ++ci)
                acc = fmaf(sWv[co * EC + ci], f2v[ci], acc);
            sV[wave][n * 16 + co] = acc;
        }
        sV[wave][n * 16 + 12] = 0.f; sV[wave][n * 16 + 13] = 0.f;
        sV[wave][n * 16 + 14] = 0.f; sV[wave][n * 16 + 15] = 0.f;
    }

    // k-range reduction across the wave (rank-1 scores: row max = q*kmax or q*kmin)
    float kmx = fmaxf(kown0, kown1);
    float kmn = fminf(kown0, kown1);
    #pragma unroll
    for (int off = 16; off > 0; off >>= 1) {
        kmx = fmaxf(kmx, __shfl_xor(kmx, off, 32));
        kmn = fminf(kmn, __shfl_xor(kmn, off, 32));
    }

    const int   halfi = lane >> 4;     // K-half of the WMMA fragment
    const int   lm    = lane & 15;     // M (for A rows) / N (for B/D cols)
    const float gam   = sScal[2];
    const float* kw = &sK[wave][0];
    const float* qw = &sQ[wave][0];
    const float* vw = &sV[wave][0];

    // ---------------- softmax(P) @ V via v_wmma_f32_16x16x4_f32 ----------------
    for (int nt = 0; nt < 4; ++nt) {                 // 16-row tiles of P
        const float qrow = qw[nt * 16 + lm];
        const float mx   = qrow * (qrow >= 0.f ? kmx : kmn);
        v8f acc = {0.f, 0.f, 0.f, 0.f, 0.f, 0.f, 0.f, 0.f};
        float ssum = 0.f;
        #pragma unroll
        for (int mt = 0; mt < 16; ++mt) {            // K-chunks of 4 tokens
            const int m0 = mt * 4 + 2 * halfi;
            const float ka = kw[m0], kb = kw[m0 + 1];
            const float pa = __expf(fmaf(qrow, ka, -mx));
            const float pb = __expf(fmaf(qrow, kb, -mx));
            ssum += pa + pb;
            v2f A;  A.x  = pa;                A.y  = pb;                 // A[M][K], K=2*halfi(+1)
            v2f Bv; Bv.x = vw[m0 * 16 + lm];  Bv.y = vw[(m0 + 1) * 16 + lm]; // B[K][N]
            acc = __builtin_amdgcn_wmma_f32_16x16x4_f32(
                false, A, false, Bv, (short)0, acc, false, false);
        }
        // softmax denominators for the 16 rows of this tile
        ssum += __shfl_xor(ssum, 16, 32);
        if (halfi == 0) sS[wave][lm] = ssum;         // same-wave DS ops are in order

        // epilogue: scale, Wp/bp, gamma-residual, scatter via upsample2 inverse
        #pragma unroll
        for (int j = 0; j < 8; ++j) {
            const int M  = j + 8 * halfi;
            const int n  = nt * 16 + M;
            const int c4 = lm;
            if (c4 < EC) {
                const float inv = 1.0f / sS[wave][M];
                float o = acc[j] * inv;
                o = fmaf(o, sWp_[c4], sbp[c4]);
                const int ty = n >> 3, tx = n & 7;
                const int c  = c4 >> 2, kk = c4 & 3;
                const int rr = baseRow + 4 * (ty >> 1) + (ty & 1) + 2 * (kk >> 1);
                const int cc = baseCol + 4 * (tx >> 1) + (tx & 1) + ((kk & 1) << 1);
                const size_t idx = ((size_t)c * HIN + rr) * WIN_ + cc;
                const float f1val = x1b[idx] + sFeat[0][c4];   // residual (L0-hot reload)
                out[(size_t)b * (CIN * HIN * WIN_) + idx] = fmaf(gam, o, f1val);
            }
        }
    }
}

extern "C" void kernel_launch(void* const* d_in, const int* in_sizes, int n_in,
                              void* d_out, int out_size, void* d_ws, size_t ws_size,
                              hipStream_t stream) {
    (void)in_sizes; (void)n_in; (void)out_size; (void)d_ws; (void)ws_size;
    const float* x1    = (const float*)d_in[0];
    const float* x2    = (const float*)d_in[1];
    const float* t1    = (const float*)d_in[2];
    const float* t2    = (const float*)d_in[3];
    const float* Wq    = (const float*)d_in[4];
    const float* bq    = (const float*)d_in[5];
    const float* Wk    = (const float*)d_in[6];
    const float* bk    = (const float*)d_in[7];
    const float* Wv    = (const float*)d_in[8];
    const float* bv    = (const float*)d_in[9];
    const float* Wp    = (const float*)d_in[10];
    const float* bp    = (const float*)d_in[11];
    const float* gamma = (const float*)d_in[12];
    const float* Wt    = (const float*)d_in[13];
    const float* bt    = (const float*)d_in[14];
    float* outp = (float*)d_out;

    const int blocks = TOTAL_WINDOWS / WAVES_PER_BLOCK;   // 2048
    tsgm_fused_kernel<<<blocks, 256, 0, stream>>>(
        x1, x2, t1, t2, Wq, bq, Wk, bk, Wv, bv, Wp, bp, gamma, Wt, bt, outp);
}